// VGGBlock2_55035710931506
// MI455X (gfx1250) — compile-verified
//
#include <hip/hip_runtime.h>
#include <hip/hip_bf16.h>

#define NSITES 500000
#define CIN    32
#define COUT   32
#define BDIM   4
#define DDIM   128
#define HDIM   128
#define WDIM   128
#define NGRID  (BDIM * DDIM * HDIM * WDIM)   // 8,388,608
#define KOFFS  125
#define RED_BLOCKS 512
#define EPSV   1e-4f

typedef float v2f __attribute__((ext_vector_type(2)));
typedef float v4f __attribute__((ext_vector_type(4)));
typedef float v8f __attribute__((ext_vector_type(8)));
typedef unsigned int v4u __attribute__((ext_vector_type(4)));
typedef int v4i __attribute__((ext_vector_type(4)));
typedef int v8i __attribute__((ext_vector_type(8)));

// ---------------------------------------------------------------- init grid
__global__ __launch_bounds__(256) void initgrid_kernel(int* __restrict__ grid) {
    long i = (long)blockIdx.x * 256 + threadIdx.x;
    if (i < NGRID) grid[i] = -1;
}

// ---------------------------------------------------------------- scatter site ids
__global__ __launch_bounds__(256) void scatter_kernel(const int* __restrict__ coords,
                                                      int* __restrict__ grid) {
    int i = blockIdx.x * 256 + threadIdx.x;
    if (i < NSITES) {
        int b = coords[i * 4 + 0];
        int z = coords[i * 4 + 1];
        int y = coords[i * 4 + 2];
        int x = coords[i * 4 + 3];
        grid[((b * DDIM + z) * HDIM + y) * WDIM + x] = i;
    }
}

// ---------------------------------------------------------------- per-channel partial sums
__global__ __launch_bounds__(256) void reduce_kernel(const float* __restrict__ f,
                                                     float* __restrict__ psum,
                                                     float* __restrict__ psumsq) {
    __shared__ float ls[8][32];
    __shared__ float ls2[8][32];
    const int lane = threadIdx.x & 31;           // channel
    const int warp = threadIdx.x >> 5;
    const int gw   = blockIdx.x * 8 + warp;
    const int nw   = RED_BLOCKS * 8;
    float s = 0.f, s2 = 0.f;
    for (int r = gw; r < NSITES; r += nw) {
        float v = f[(long)r * CIN + lane];
        s  += v;
        s2 += v * v;
    }
    ls[warp][lane] = s;
    ls2[warp][lane] = s2;
    __syncthreads();
    if (warp == 0) {
        float ts = 0.f, ts2 = 0.f;
        #pragma unroll
        for (int w = 0; w < 8; ++w) { ts += ls[w][lane]; ts2 += ls2[w][lane]; }
        psum[blockIdx.x * 32 + lane]   = ts;
        psumsq[blockIdx.x * 32 + lane] = ts2;
    }
}

// ---------------------------------------------------------------- finalize scale/shift
__global__ void finalize_kernel(const float* __restrict__ psum,
                                const float* __restrict__ psumsq,
                                const float* __restrict__ gamma,
                                const float* __restrict__ beta,
                                float* __restrict__ scale,
                                float* __restrict__ shift) {
    int c = threadIdx.x;                          // 32 threads
    float s = 0.f, s2 = 0.f;
    for (int b = 0; b < RED_BLOCKS; ++b) {        // fixed order -> deterministic
        s  += psum[b * 32 + c];
        s2 += psumsq[b * 32 + c];
    }
    float mean = s / (float)NSITES;
    float var  = s2 / (float)NSITES - mean * mean;
    float rstd = rsqrtf(var + EPSV);
    float sc   = gamma[c] * rstd;
    scale[c] = sc;
    shift[c] = beta[c] - mean * sc;
}

// ---------------------------------------------------------------- fused BN + ReLU
__global__ __launch_bounds__(256) void bnrelu_kernel(const float* __restrict__ f,
                                                     const float* __restrict__ scale,
                                                     const float* __restrict__ shift,
                                                     float* __restrict__ y) {
    long i = (long)blockIdx.x * 256 + threadIdx.x;     // float4 index
    const long n4 = (long)NSITES * CIN / 4;
    if (i >= n4) return;
    float4 v  = ((const float4*)f)[i];
    int cb    = (int)((i & 7) << 2);
    float4 r;
    r.x = fmaxf(0.f, v.x * scale[cb + 0] + shift[cb + 0]);
    r.y = fmaxf(0.f, v.y * scale[cb + 1] + shift[cb + 1]);
    r.z = fmaxf(0.f, v.z * scale[cb + 2] + shift[cb + 2]);
    r.w = fmaxf(0.f, v.w * scale[cb + 3] + shift[cb + 3]);
    ((float4*)y)[i] = r;
}

// ---------------------------------------------------------------- weight pre-swizzle
// K remap: WMMA chunk kc consumes global K = 16*half + 2*kc + j  (half = lane>=16, j = vgpr).
// Per (kc, half, n0) store contiguous float4 {W[kr][n0], W[kr+1][n0], W[kr][n0+16], W[kr+1][n0+16]}
// so the conv inner loop does one conflict-free ds_load_b128 per kc.
__global__ __launch_bounds__(256) void wswz_kernel(const float* __restrict__ weight,
                                                   float* __restrict__ wswz) {
    const int o    = blockIdx.x;                 // kernel offset 0..124
    const int t    = threadIdx.x;                // 256 = 8 kc * 2 half * 16 n0
    const int kc   = t >> 5;
    const int half = (t >> 4) & 1;
    const int n0   = t & 15;
    const int kr   = half * 16 + kc * 2;
    const float* wb = weight + o * (CIN * COUT);
    v4f q = { wb[kr * COUT + n0],      wb[(kr + 1) * COUT + n0],
              wb[kr * COUT + n0 + 16], wb[(kr + 1) * COUT + n0 + 16] };
    ((v4f*)wswz)[o * 256 + t] = q;
}

// ---------------------------------------------------------------- TDM: 4KB contiguous copy
__device__ __forceinline__ void tdm_load_4k(const float* gsrc, unsigned lds_off) {
    unsigned long long ga = (unsigned long long)(size_t)gsrc;
    // Group0: count=1 | lds_addr | global_addr[56:0] | type=2
    v4u g0 = { 1u, lds_off, (unsigned)ga,
               ((unsigned)((ga >> 32) & 0x01FFFFFFull)) | 0x80000000u };
    // Group1: data_size=4B; tensor_dim0=1024, tensor_dim1=1, tile_dim0=1024,
    //         tile_dim1=1, tile_dim2=0, tensor_dim0_stride=1024
    v8i g1 = { 0x00020000,
               (int)(1024u << 16),           // tensor_dim0[15:0] << 16
               (int)(1u << 16),              // tensor_dim0[31:16]=0 | tensor_dim1[15:0]<<16
               (int)(1024u << 16),           // tensor_dim1[31:16]=0 | tile_dim0<<16
               1,                            // tile_dim1=1 | tile_dim2=0
               1024,                         // tensor_dim0_stride[31:0]
               0, 0 };
    v4i g2 = { 1, 1, 0, 0 };                 // tensor_dim2=1, tensor_dim3=1 (unused tiles)
    v4i g3 = { 0, 0, 0, 0 };
#if __clang_major__ >= 23
    v8i g4 = { 0, 0, 0, 0, 0, 0, 0, 0 };
    __builtin_amdgcn_tensor_load_to_lds(g0, g1, g2, g3, g4, 0);
#else
    __builtin_amdgcn_tensor_load_to_lds(g0, g1, g2, g3, 0);
#endif
}

// ---------------------------------------------------------------- WMMA sparse conv
// One wave = 16 output sites. Per offset: 8 x ds_load_b128 (B), 4 x global b128 gather (A),
// 16 x v_wmma_f32_16x16x4_f32. Weights double-buffered via tensor_load_to_lds.
__global__ __launch_bounds__(256) void conv_kernel(const float* __restrict__ yfeat,
                                                   const int* __restrict__ coords,
                                                   const float* __restrict__ wswz,
                                                   const int* __restrict__ grid,
                                                   float* __restrict__ out) {
    __shared__ v4f ldsB[2][256];                       // 2 x 4 KB weight buffers
    const int lane      = threadIdx.x & 31;
    const int wave      = threadIdx.x >> 5;
    const int tile_base = (blockIdx.x * 8 + wave) * 16;
    const int row       = lane & 15;                   // lanes l and l+16 mirror the same row
    const int site      = tile_base + row;
    const bool site_ok  = site < NSITES;
    const int half      = lane >> 4;                   // K-half select
    const int n0        = lane & 15;

    int cb = 0, cz = 0, cy = 0, cx = 0;
    if (site_ok) {
        const int4 cc = ((const int4*)coords)[site];
        cb = cc.x; cz = cc.y; cy = cc.z; cx = cc.w;
    }

    v8f c0 = {};
    v8f c1 = {};

    if (wave == 0)                                     // prefetch offset 0 into buffer 0
        tdm_load_4k(wswz, (unsigned)(size_t)&ldsB[0][0]);

    for (int o = 0; o < KOFFS; ++o) {
        __syncthreads();                               // prev compute done -> other buf free
        if (wave == 0) {
            if (o + 1 < KOFFS) {
                tdm_load_4k(wswz + (size_t)(o + 1) * (CIN * COUT),
                            (unsigned)(size_t)&ldsB[(o + 1) & 1][0]);
                __builtin_amdgcn_s_wait_tensorcnt((short)1);   // oldest (offset o) done
            } else {
                __builtin_amdgcn_s_wait_tensorcnt((short)0);
            }
        }
        __syncthreads();                               // buf[o&1] visible to all waves

        const int dz = o / 25 - 2;
        const int dy = (o / 5) % 5 - 2;
        const int dx = o % 5 - 2;
        const int nz = cz + dz, ny = cy + dy, nx = cx + dx;
        const bool inb = site_ok && nz >= 0 && nz < DDIM && ny >= 0 && ny < HDIM &&
                         nx >= 0 && nx < WDIM;
        int nidx = -1;
        if (inb) nidx = grid[((cb * DDIM + nz) * HDIM + ny) * WDIM + nx];
        const bool valid = nidx >= 0;

        // Wave-uniform skip (EXEC stays all-1s around the WMMAs).
        if (__builtin_amdgcn_ballot_w32(valid) == 0u) continue;

        // A gather: lane-half h reads 16 contiguous floats y[nrow][16h .. 16h+15].
        const v4f* srcq = (const v4f*)(yfeat + (size_t)nidx * CIN) + half * 4;
        v4f aq[4];
        #pragma unroll
        for (int i = 0; i < 4; ++i) {
            v4f z = { 0.f, 0.f, 0.f, 0.f };
            aq[i] = valid ? srcq[i] : z;
        }

        const v4f* bb = &ldsB[o & 1][(half << 4) + n0];
        #pragma unroll
        for (int kc = 0; kc < 8; ++kc) {
            v4f bq = bb[kc << 5];                      // one ds_load_b128, conflict-free
            v2f a  = (kc & 1) ? aq[kc >> 1].zw : aq[kc >> 1].xy;
            c0 = __builtin_amdgcn_wmma_f32_16x16x4_f32(false, a, false, bq.xy,
                                                       (short)0, c0, false, false);
            c1 = __builtin_amdgcn_wmma_f32_16x16x4_f32(false, a, false, bq.zw,
                                                       (short)0, c1, false, false);
        }
    }

    // Store: VGPR v -> row v + 8*half; col = n0 (+16 for c1).
    #pragma unroll
    for (int v = 0; v < 8; ++v) {
        const int srow = tile_base + v + 8 * half;
        if (srow < NSITES) {
            out[(long)srow * COUT + n0]      = c0[v];
            out[(long)srow * COUT + n0 + 16] = c1[v];
        }
    }
}

// ---------------------------------------------------------------- launch
extern "C" void kernel_launch(void* const* d_in, const int* in_sizes, int n_in,
                              void* d_out, int out_size, void* d_ws, size_t ws_size,
                              hipStream_t stream) {
    const float* features = (const float*)d_in[0];
    const int*   coords   = (const int*)d_in[1];
    const float* gamma    = (const float*)d_in[2];
    const float* beta     = (const float*)d_in[3];
    const float* weight   = (const float*)d_in[4];
    float*       out      = (float*)d_out;

    char*  ws     = (char*)d_ws;
    int*   grid   = (int*)ws;                                    // 33.5 MB
    float* yfeat  = (float*)(ws + (size_t)NGRID * sizeof(int));  // 64 MB
    float* psum   = yfeat + (size_t)NSITES * CIN;
    float* psumsq = psum + RED_BLOCKS * 32;
    float* scale  = psumsq + RED_BLOCKS * 32;
    float* shift  = scale + 32;
    float* wswz   = shift + 32;                                  // 512 KB, 16B-aligned

    initgrid_kernel<<<(NGRID + 255) / 256, 256, 0, stream>>>(grid);
    scatter_kernel<<<(NSITES + 255) / 256, 256, 0, stream>>>(coords, grid);
    reduce_kernel<<<RED_BLOCKS, 256, 0, stream>>>(features, psum, psumsq);
    finalize_kernel<<<1, 32, 0, stream>>>(psum, psumsq, gamma, beta, scale, shift);
    const long n4 = (long)NSITES * CIN / 4;
    bnrelu_kernel<<<(int)((n4 + 255) / 256), 256, 0, stream>>>(features, scale, shift, yfeat);
    wswz_kernel<<<KOFFS, 256, 0, stream>>>(weight, wswz);
    const int tiles = (NSITES + 127) / 128;   // 8 waves/block * 16 rows/wave
    conv_kernel<<<tiles, 256, 0, stream>>>(yfeat, coords, wswz, grid, out);
}